// DTCWTLayerBiort_bpInv_30253749633564
// MI455X (gfx1250) — compile-verified
//
#include <hip/hip_runtime.h>

// DTCWT inverse level, fully fused for MI455X (gfx1250, wave32).
//
// Roofline: 670 MB compulsory HBM traffic vs 6.9 GFLOP useful -> bandwidth
// bound (~29 us floor at 23.3 TB/s). Single fused kernel (c2q + col filters +
// row filters) so HBM is touched exactly once; all intermediates live in the
// 320 KB WGP LDS. The separable convs are expressed as banded matmuls on
// V_WMMA_F32_16X16X4_F32 (fp32 data => fp32 WMMA), keeping VALU free.
//
// Tiling: one workgroup (256 thr = 8 wave32) per 64x64 output tile of one
// (b,c) plane. Staged-in-LDS region spans cols [c0-16, c0+80) (96 cols) and
// rows with per-filter halos. Intermediates t,u,v (64x96) also in LDS.

typedef float v2f __attribute__((ext_vector_type(2)));
typedef float v8f __attribute__((ext_vector_type(8)));

// ---- geometry ----
#define HW   256
#define HHW  128
#define S_IN 104   // LDS row stride for staged inputs  (bank-conflict-free B-chunk reads)
#define S_MID 100  // LDS row stride for intermediates  (bank-conflict-free A-chunk reads)

// LDS layout (float offsets)
#define OFF_B0 0        // band g0: 28x16
#define OFF_B1 448      // band g1: 36x16 (J=34 padded to 36 with zero rows)
#define OFF_B2 1024     // band g2: 28x16
#define OFF_YL 1472     // Yl tile: 76 x 96  (stride 104)
#define OFF_LH 9376     // lh tile: 84 x 96  (stride 104; 82 needed + 2 pad rows for J=36 chunks)
#define OFF_HL 18112    // hl tile: 76 x 96
#define OFF_HH 26016    // hh tile: 76 x 96
#define OFF_T  33920    // t: 64 x 96 (stride 100)
#define OFF_U  40320    // u
#define OFF_V  46720    // v
#define LDS_FLOATS 53120  // 212,480 B  (< 320 KB WGP LDS)

__device__ __forceinline__ int reflect256(int i) {
  // symmetric padding: -1 -> 0, -2 -> 1, N -> N-1, ... (halo <= 16, single reflection)
  if (i < 0)    i = -1 - i;
  if (i >= HW)  i = 2 * HW - 1 - i;
  return i;
}

__device__ __forceinline__ v8f wmma4(v2f a, v2f b, v8f c) {
  // D(16x16,f32) = A(16x4,f32) x B(4x16,f32) + C
  return __builtin_amdgcn_wmma_f32_16x16x4_f32(
      /*neg_a=*/false, a, /*neg_b=*/false, b,
      /*c_mod=*/(short)0, c, /*reuse_a=*/false, /*reuse_b=*/false);
}

// Band matrix chunk, usable both as A-role (col pass: A[m][k]=band[k0+k][m])
// and as B-role (row pass: B[k][n]=band[k0+k][n]) -- identical fetch pattern
// under the ISA 32-bit A / B VGPR layouts (K split 0,2 / 1,3 across lane halves).
__device__ __forceinline__ v2f band_pair(const float* band, int k0, int lane) {
  int kk = k0 + ((lane & 16) ? 2 : 0);
  int m  = lane & 15;
  v2f r;
  r.x = band[kk * 16 + m];
  r.y = band[(kk + 1) * 16 + m];
  return r;
}

// B-role data chunk for the column pass: B[k][n] = X[row0 + k0 + k][col0 + n]
__device__ __forceinline__ v2f data_colB(const float* x, int xs, int row0, int col0,
                                         int k0, int lane) {
  int kk = k0 + ((lane & 16) ? 2 : 0);
  int n  = lane & 15;
  v2f r;
  r.x = x[(row0 + kk)     * xs + col0 + n];
  r.y = x[(row0 + kk + 1) * xs + col0 + n];
  return r;
}

// A-role data chunk for the row pass: A[m][k] = X[row0 + m][cbase + k0 + k]
__device__ __forceinline__ v2f data_rowA(const float* x, int xs, int row0, int cbase,
                                         int k0, int lane) {
  int kk = k0 + ((lane & 16) ? 2 : 0);
  int m  = lane & 15;
  v2f r;
  r.x = x[(row0 + m) * xs + cbase + kk];
  r.y = x[(row0 + m) * xs + cbase + kk + 1];
  return r;
}

__device__ __forceinline__ v8f col_chain(v8f acc, const float* band, int J,
                                         const float* x, int r0t, int c0t, int lane) {
  for (int k0 = 0; k0 < J; k0 += 4) {
    v2f a = band_pair(band, k0, lane);
    v2f b = data_colB(x, S_IN, r0t, c0t, k0, lane);
    acc = wmma4(a, b, acc);
  }
  return acc;
}

__device__ __forceinline__ v8f row_chain(v8f acc, const float* x, int r0t, int cbase,
                                         const float* band, int J, int lane) {
  for (int k0 = 0; k0 < J; k0 += 4) {
    v2f a = data_rowA(x, S_MID, r0t, cbase, k0, lane);
    v2f b = band_pair(band, k0, lane);
    acc = wmma4(a, b, acc);
  }
  return acc;
}

// Scatter a 16x16 f32 C/D tile (VGPR i: lanes0-15 -> M=i, lanes16-31 -> M=i+8)
__device__ __forceinline__ void store_tile_lds(float* dst, int r0t, int c0t,
                                               v8f d, int lane) {
  int n     = lane & 15;
  int rbase = r0t + ((lane & 16) ? 8 : 0);
#pragma unroll
  for (int i = 0; i < 8; ++i)
    dst[(rbase + i) * S_MID + c0t + n] = d[i];
}

__device__ __forceinline__ void stage_band(float* dst, const float* g, int L, int J,
                                           int tid) {
  // band[j][i] = flip(g)[j-i]  (zero outside [0,L)); padded rows j>=15+L are zero.
  for (int idx = tid; idx < J * 16; idx += 256) {
    int j = idx / 16, i = idx % 16;
    int u = j - i;
    dst[idx] = (u >= 0 && u < L) ? g[L - 1 - u] : 0.0f;
  }
}

// c2q: subband pair (s1,s2) of (Yhr,Yhi) -> real quad image tile, scaled 1/sqrt2.
__device__ __forceinline__ void stage_c2q(float* dst, int rows,
                                          const float* hr1, const float* hi1,
                                          const float* hr2, const float* hi2,
                                          int rtop, int cleft, int tid) {
  const float is2 = 0.70710678118654752440f;
  for (int idx = tid; idx < rows * 96; idx += 256) {
    int rr = idx / 96, cc = idx % 96;
    int R = reflect256(rtop + rr);
    int C = reflect256(cleft + cc);
    int o = (R >> 1) * HHW + (C >> 1);
    float v;
    if ((R & 1) == 0) v = ((C & 1) == 0) ? (hr1[o] + hr2[o]) : (hi1[o] + hi2[o]);
    else              v = ((C & 1) == 0) ? (hi1[o] - hi2[o]) : (hr2[o] - hr1[o]);
    dst[rr * S_IN + cc] = v * is2;
  }
}

__global__ void dtcwt_inv_fused(const float* __restrict__ Yl,
                                const float* __restrict__ Yhr,
                                const float* __restrict__ Yhi,
                                const float* __restrict__ g0,
                                const float* __restrict__ g1,
                                const float* __restrict__ g2,
                                float* __restrict__ out) {
  extern __shared__ float smem[];
  const int tid   = threadIdx.x;
  const int lane  = tid & 31;
  const int wave  = tid >> 5;

  const int plane = blockIdx.x >> 4;          // 512 planes (b*64 + c)
  const int tile  = blockIdx.x & 15;          // 4x4 tiles of 64x64
  const int r0    = (tile >> 2) << 6;
  const int c0    = (tile & 3)  << 6;

  const float* Ylp = Yl + (size_t)plane * (HW * HW);
  const size_t sb  = (size_t)plane * 6 * (HHW * HHW);
  const float* hr  = Yhr + sb;
  const float* hi  = Yhi + sb;
  const int SBN    = HHW * HHW;

  float* bnd0 = smem + OFF_B0;
  float* bnd1 = smem + OFF_B1;
  float* bnd2 = smem + OFF_B2;
  float* sYl  = smem + OFF_YL;
  float* sLh  = smem + OFF_LH;
  float* sHl  = smem + OFF_HL;
  float* sHh  = smem + OFF_HH;
  float* sT   = smem + OFF_T;
  float* sU   = smem + OFF_U;
  float* sV   = smem + OFF_V;

  // ---- stage taps (banded matrices) and input tiles into LDS ----
  stage_band(bnd0, g0, 13, 28, tid);
  stage_band(bnd1, g1, 19, 36, tid);
  stage_band(bnd2, g2, 13, 28, tid);

  for (int idx = tid; idx < 76 * 96; idx += 256) {  // Yl halo +-6 rows, 96 cols
    int rr = idx / 96, cc = idx % 96;
    sYl[rr * S_IN + cc] =
        Ylp[reflect256(r0 - 6 + rr) * HW + reflect256(c0 - 16 + cc)];
  }
  // lh = c2q(subbands 0,5): halo +-9 (+2 pad rows for padded-J reads)
  stage_c2q(sLh, 84, hr + 0 * SBN, hi + 0 * SBN, hr + 5 * SBN, hi + 5 * SBN,
            r0 - 9, c0 - 16, tid);
  // hl = c2q(2,3): halo +-6
  stage_c2q(sHl, 76, hr + 2 * SBN, hi + 2 * SBN, hr + 3 * SBN, hi + 3 * SBN,
            r0 - 6, c0 - 16, tid);
  // hh = c2q(1,4): halo +-6
  stage_c2q(sHh, 76, hr + 1 * SBN, hi + 1 * SBN, hr + 4 * SBN, hi + 4 * SBN,
            r0 - 6, c0 - 16, tid);
  __syncthreads();

  // ---- phase 1: column filters -> t,u,v (64 rows x 96 cols) via WMMA chains ----
  // 72 tile jobs: kind 0: t = g0*Yl + g1*lh (16 wmma), 1: u = g0*hl (7), 2: v = g2*hh (7)
  for (int job = wave; job < 72; job += 8) {
    int kind = job / 24;
    int sub  = job % 24;
    int rt16 = (sub / 6) << 4;
    int ct16 = (sub % 6) << 4;
    v8f acc = {0.f, 0.f, 0.f, 0.f, 0.f, 0.f, 0.f, 0.f};
    if (kind == 0) {
      acc = col_chain(acc, bnd0, 28, sYl, rt16, ct16, lane);
      acc = col_chain(acc, bnd1, 36, sLh, rt16, ct16, lane);
      store_tile_lds(sT, rt16, ct16, acc, lane);
    } else if (kind == 1) {
      acc = col_chain(acc, bnd0, 28, sHl, rt16, ct16, lane);
      store_tile_lds(sU, rt16, ct16, acc, lane);
    } else {
      acc = col_chain(acc, bnd2, 28, sHh, rt16, ct16, lane);
      store_tile_lds(sV, rt16, ct16, acc, lane);
    }
  }
  __syncthreads();

  // ---- phase 2: row filters, accumulate all three, store to HBM ----
  // intermediate col index of output col (ct16+c): ct16 + 16 - m + j
  for (int job = wave; job < 16; job += 8) {
    int rt16 = (job >> 2) << 4;
    int ct16 = (job & 3)  << 4;
    v8f acc = {0.f, 0.f, 0.f, 0.f, 0.f, 0.f, 0.f, 0.f};
    acc = row_chain(acc, sT, rt16, ct16 + 10, bnd0, 28, lane);  // g0, m=6
    acc = row_chain(acc, sU, rt16, ct16 + 7,  bnd1, 36, lane);  // g1, m=9
    acc = row_chain(acc, sV, rt16, ct16 + 10, bnd2, 28, lane);  // g2, m=6

    float* op  = out + (size_t)plane * (HW * HW);
    int n      = lane & 15;
    int rbase  = r0 + rt16 + ((lane & 16) ? 8 : 0);
    int cbase  = c0 + ct16 + n;
#pragma unroll
    for (int i = 0; i < 8; ++i)
      op[(rbase + i) * HW + cbase] = acc[i];
  }
}

extern "C" void kernel_launch(void* const* d_in, const int* in_sizes, int n_in,
                              void* d_out, int out_size, void* d_ws, size_t ws_size,
                              hipStream_t stream) {
  (void)in_sizes; (void)n_in; (void)d_ws; (void)ws_size; (void)out_size;
  const float* Yl  = (const float*)d_in[0];
  const float* Yhr = (const float*)d_in[1];
  const float* Yhi = (const float*)d_in[2];
  const float* g0  = (const float*)d_in[3];
  const float* g1  = (const float*)d_in[4];
  const float* g2  = (const float*)d_in[5];
  float* out = (float*)d_out;

  // 512 planes x 16 (64x64) tiles; 256 threads = 8 wave32 per block.
  dim3 grid(512 * 16);
  dim3 block(256);
  size_t smem = (size_t)LDS_FLOATS * sizeof(float);  // ~208 KB of the 320 KB WGP LDS
  hipLaunchKernelGGL(dtcwt_inv_fused, grid, block, smem, stream,
                     Yl, Yhr, Yhi, g0, g1, g2, out);
}